// NeuralIDE_wODE_42511586296394
// MI455X (gfx1250) — compile-verified
//
#include <hip/hip_runtime.h>
#include <hip/hip_bf16.h>
#include <math.h>

// ---------------------------------------------------------------------------
// Neural IDE solver, refactored to avoid materializing the 268MB K tensor:
//   G = (w ⊙_j hK) @ M + w @ (fy @ bk2^T),  M = fy @ Wk2^T (viewed [T*HK, D])
// All contractions run as fp32 WMMA (v_wmma_f32_16x16x4_f32), exact precision.
// Inner GEMM loops unrolled x4 (K step 16) with two interleaved accumulators
// to keep the matrix pipe fed from a single wave.
// ---------------------------------------------------------------------------

#define T_  128
#define D_  64
#define H_  256
#define HK_ 128
#define R_  (HK_ * D_)      // 8192
#define KBIG_ (T_ * HK_)    // 16384
#define NSPLIT_ 32
#define KCHUNK_ (KBIG_ / NSPLIT_)  // 512
#define MAXIT_ 3

typedef __attribute__((ext_vector_type(2))) float v2f;
typedef __attribute__((ext_vector_type(8))) float v8f;

__device__ __forceinline__ v8f wmma4(v2f a, v2f b, v8f c) {
    // D = A(16x4 f32) * B(4x16 f32) + C(16x16 f32)
    return __builtin_amdgcn_wmma_f32_16x16x4_f32(
        /*neg_a=*/false, a, /*neg_b=*/false, b,
        /*c_mod=*/(short)0, c, /*reuse_a=*/false, /*reuse_b=*/false);
}

// ---------------------------------------------------------------------------
// Generic 16x16-tile fp32 WMMA GEMM: C = epilogue(A[M,K] * B, partial K range)
//   TRANSB=false: B row-major [K,N];  TRANSB=true: B stored [N,K] row-major.
//   EP: 0 = plain store, 1 = tanh(c + bias[n]), 2 = c + bias[n]
//   K-split: blockIdx.y picks k-range [y*kChunk, (y+1)*kChunk), output offset
//   y*partStride (partStride=0 & gridDim.y=1 for unsplit calls).
//   One wave (32 lanes) per tile; 4 waves per block; wave32 EXEC all-ones.
//   REQUIRES: kChunk (and K) multiples of 16 — true for every call site here.
// ---------------------------------------------------------------------------
template<int EP, bool TRANSB>
__global__ void gemm16(const float* __restrict__ A, const float* __restrict__ Bm,
                       float* __restrict__ Cout, const float* __restrict__ bias,
                       int M, int N, int K, int kChunk, int partStride)
{
    const int wave = threadIdx.x >> 5;
    const int lane = threadIdx.x & 31;
    const int tilesN = N >> 4;
    const int numTiles = (M >> 4) * tilesN;
    const int tile = blockIdx.x * 4 + wave;
    if (tile >= numTiles) return;                  // wave-uniform exit

    const int tm   = (tile / tilesN) << 4;
    const int tn   = (tile % tilesN) << 4;
    const int half = lane >> 4;                    // 0: K=0,1 | 1: K=2,3
    const int l16  = lane & 15;
    const int arow = tm + l16;
    const int bcol = tn + l16;

    const int kBegin = blockIdx.y * kChunk;
    int kEnd = kBegin + kChunk; if (kEnd > K) kEnd = K;
    float* __restrict__ Cp = Cout + (size_t)blockIdx.y * (size_t)partStride;

    const float* __restrict__ Arow = A + (size_t)arow * K;
    const float* __restrict__ Bcol = TRANSB ? (Bm + (size_t)bcol * K) : Bm;

    v8f c0 = {}, c1 = {};
    for (int k = kBegin; k < kEnd; k += 16) {
        const int ka = k + half * 2;               // lane's K base within step
        // A fragments: 16x4 f32 layout (lanes 0-15: K=0,1 | lanes 16-31: K=2,3)
        v2f a0 = *(const v2f*)(Arow + ka);
        v2f a1 = *(const v2f*)(Arow + ka + 4);
        v2f a2 = *(const v2f*)(Arow + ka + 8);
        v2f a3 = *(const v2f*)(Arow + ka + 12);
        v2f b0, b1, b2, b3;
        if (TRANSB) {
            b0 = *(const v2f*)(Bcol + ka);
            b1 = *(const v2f*)(Bcol + ka + 4);
            b2 = *(const v2f*)(Bcol + ka + 8);
            b3 = *(const v2f*)(Bcol + ka + 12);
        } else {
            b0.x = Bm[(size_t)(ka +  0) * N + bcol];
            b0.y = Bm[(size_t)(ka +  1) * N + bcol];
            b1.x = Bm[(size_t)(ka +  4) * N + bcol];
            b1.y = Bm[(size_t)(ka +  5) * N + bcol];
            b2.x = Bm[(size_t)(ka +  8) * N + bcol];
            b2.y = Bm[(size_t)(ka +  9) * N + bcol];
            b3.x = Bm[(size_t)(ka + 12) * N + bcol];
            b3.y = Bm[(size_t)(ka + 13) * N + bcol];
        }
        // Two independent accumulation chains keep the XDL pipe busy.
        c0 = wmma4(a0, b0, c0);
        c1 = wmma4(a1, b1, c1);
        c0 = wmma4(a2, b2, c0);
        c1 = wmma4(a3, b3, c1);
    }

    // C/D layout: VGPR v -> row tm+v (lanes 0-15) / tm+v+8 (lanes 16-31)
#pragma unroll
    for (int v = 0; v < 8; ++v) {
        const int row = tm + v + half * 8;
        float val = c0[v] + c1[v];
        if (EP == 1) val = tanhf(val + bias[bcol]);
        if (EP == 2) val = val + bias[bcol];
        Cp[(size_t)row * N + bcol] = val;
    }
}

// Trapezoid weights w[T,T] and fixed-point init y[i,:] = z0.
__global__ void build_w(const float* __restrict__ t, float* __restrict__ w,
                        float* __restrict__ y, const float* __restrict__ z0)
{
    const int idx = blockIdx.x * blockDim.x + threadIdx.x;  // T*T
    const int i = idx >> 7, j = idx & 127;
    const float dt = t[1] - t[0];
    float val = (j <= i) ? dt : 0.0f;
    if (j == i) val *= 0.5f;
    if (j == 0) val *= 0.5f;
    if (i == 0) val = 0.0f;
    w[idx] = val;
    if (j < D_) y[i * D_ + j] = z0[j];
}

// hKw[i, j*HK + h] = w[i,j] * tanh(t_i*Wk1[0,h] + t_j*Wk1[1,h] + bk1[h])
// (iteration-invariant; 8MB, stays L2-resident)
__global__ void build_hKw(const float* __restrict__ t, const float* __restrict__ Wk1,
                          const float* __restrict__ bk1, const float* __restrict__ w,
                          float* __restrict__ hKw)
{
    const int i = blockIdx.x >> 7, j = blockIdx.x & 127;
    const int h = threadIdx.x;
    const float v = tanhf(t[i] * Wk1[h] + t[j] * Wk1[HK_ + h] + bk1[h]);
    hKw[((size_t)i * T_ + j) * HK_ + h] = w[i * T_ + j] * v;
}

// G = sum of NSPLIT K-split partials + (w @ bt) term.
__global__ void reduce_G(const float* __restrict__ Gp, const float* __restrict__ wbt,
                         float* __restrict__ G, int nparts)
{
    const int idx = blockIdx.x * blockDim.x + threadIdx.x;  // T*D
    float s = wbt[idx];
    for (int p = 0; p < nparts; ++p) s += Gp[(size_t)p * (T_ * D_) + idx];
    G[idx] = s;
}

__global__ void final_out(const float* __restrict__ y, float* __restrict__ out)
{
    const int d = threadIdx.x;  // 64
    out[d] = y[(T_ - 1) * D_ + d];
}

extern "C" void kernel_launch(void* const* d_in, const int* in_sizes, int n_in,
                              void* d_out, int out_size, void* d_ws, size_t ws_size,
                              hipStream_t stream)
{
    const float* z0  = (const float*)d_in[0];   // [1,64]
    const float* t   = (const float*)d_in[1];   // [128]
    const float* W1  = (const float*)d_in[2];   // [64,256]
    const float* b1  = (const float*)d_in[3];   // [256]
    const float* W2  = (const float*)d_in[4];   // [256,64]
    const float* b2  = (const float*)d_in[5];   // [64]
    const float* Wk1 = (const float*)d_in[6];   // [2,128]
    const float* bk1 = (const float*)d_in[7];   // [128]
    const float* Wk2 = (const float*)d_in[8];   // [128,4096] == [8192,64] rm
    const float* bk2 = (const float*)d_in[9];   // [4096]     == [64,64] rm
    float* out = (float*)d_out;

    float* ws  = (float*)d_ws;
    float* w   = ws;  ws += T_ * T_;            // 64 KB
    float* hKw = ws;  ws += (size_t)T_ * T_ * HK_;  // 8 MB
    float* y   = ws;  ws += T_ * D_;
    float* h1  = ws;  ws += T_ * H_;
    float* fy  = ws;  ws += T_ * D_;
    float* bt  = ws;  ws += T_ * D_;
    float* Mhd = ws;  ws += (size_t)T_ * R_;        // 4 MB, == Mr[16384,64] rm
    float* wbt = ws;  ws += T_ * D_;
    float* Gp  = ws;  ws += (size_t)NSPLIT_ * T_ * D_;  // 1 MB
    float* G   = ws;  ws += T_ * D_;

    build_w  <<<(T_ * T_) / 256, 256, 0, stream>>>(t, w, y, z0);
    build_hKw<<<T_ * T_, HK_, 0, stream>>>(t, Wk1, bk1, w, hKw);

    for (int it = 0; it < MAXIT_; ++it) {
        // h1 = tanh(y @ W1 + b1)        [128,64]x[64,256]
        gemm16<1, false><<<dim3((T_/16)*(H_/16)/4, 1), 128, 0, stream>>>(
            y, W1, h1, b1, T_, H_, D_, D_, 0);
        // fy = h1 @ W2 + b2             [128,256]x[256,64]
        gemm16<2, false><<<dim3((T_/16)*(D_/16)/4, 1), 128, 0, stream>>>(
            h1, W2, fy, b2, T_, D_, H_, H_, 0);
        // Mhd[j, h*64+d] = fy @ Wk2^T   [128,64]x[64,8192]  (B stored [8192,64])
        gemm16<0, true><<<dim3((T_/16)*(R_/16)/4, 1), 128, 0, stream>>>(
            fy, Wk2, Mhd, nullptr, T_, R_, D_, D_, 0);
        // bt = fy @ bk2^T               [128,64]x[64,64]    (B stored [64,64])
        gemm16<0, true><<<dim3((T_/16)*(D_/16)/4, 1), 128, 0, stream>>>(
            fy, bk2, bt, nullptr, T_, D_, D_, D_, 0);
        // wbt = w @ bt                  [128,128]x[128,64]
        gemm16<0, false><<<dim3((T_/16)*(D_/16)/4, 1), 128, 0, stream>>>(
            w, bt, wbt, nullptr, T_, D_, T_, T_, 0);
        // Gp[p] = hKw @ Mr (K-split)    [128,16384]x[16384,64]
        gemm16<0, false><<<dim3((T_/16)*(D_/16)/4, NSPLIT_), 128, 0, stream>>>(
            hKw, Mhd, Gp, nullptr, T_, D_, KBIG_, KCHUNK_, T_ * D_);
        // G = sum(Gp) + wbt
        reduce_G<<<(T_ * D_) / 256, 256, 0, stream>>>(Gp, wbt, G, NSPLIT_);
        // y = y0 + w @ G                [128,128]x[128,64], bias = z0 broadcast
        gemm16<2, false><<<dim3((T_/16)*(D_/16)/4, 1), 128, 0, stream>>>(
            w, G, y, z0, T_, D_, T_, T_, 0);
    }

    final_out<<<1, D_, 0, stream>>>(y, out);
}